// PointerGenerator_14413910245874
// MI455X (gfx1250) — compile-verified
//
#include <hip/hip_runtime.h>
#include <math.h>

// ---------------- problem dims ----------------
#define E_DIM   256
#define H_DIM   512
#define F_DIM   100
#define MCL_DIM 500
#define V_DIM   128000
#define CTX_DIM 499
#define MPAD    512          // MCL padded to multiple of 16 for WMMA tiles
#define EPS_F   1e-8f

// ---------------- ws layout (float offsets) ----------------
#define WS_FACTS   0                         // [MPAD x 256]  facts (row 499 = mean, 500..511 = 0)
#define WS_W1P     (WS_FACTS + MPAD*E_DIM)   // [MPAD x 256]  attn1_W[:, :256] padded
#define WS_HTERM   (WS_W1P   + MPAD*E_DIM)   // [MPAD]        b + h0 . attn1_W[:,256:]
#define WS_W2P     (WS_HTERM + MPAD)         // [MPAD]        attn2_W padded
#define WS_ENERGY  (WS_W2P   + MPAD)         // [MPAD]
#define WS_ATTNW   (WS_ENERGY+ MPAD)         // [256]
#define WS_X       (WS_ATTNW + E_DIM)        // [640] (612 used)
#define WS_GI      (WS_X     + 640)          // [1536]
#define WS_GH      (WS_GI    + 3*H_DIM)      // [1536]
#define WS_OUTV    (WS_GH    + 3*H_DIM)      // [768] = [h_new | attn_weighted]
#define WS_OUT1    (WS_OUTV  + 768)          // [512]

typedef __attribute__((ext_vector_type(2))) float v2f;
typedef __attribute__((ext_vector_type(8))) float v8f;

// ---------------- prep: pad W1/W2, zero energy, pad facts rows, zero dec-pos part of x ----
__global__ void k_prep(const float* __restrict__ attn1_W, const float* __restrict__ attn2_W,
                       float* __restrict__ ws) {
    int i = blockIdx.x * blockDim.x + threadIdx.x;        // 0 .. MPAD*256-1
    int j = i >> 8, k = i & 255;
    ws[WS_W1P + i] = (j < MCL_DIM) ? attn1_W[j * (E_DIM + H_DIM) + k] : 0.0f;
    if (i < MPAD)      ws[WS_W2P + i]    = (i < MCL_DIM) ? attn2_W[i] : 0.0f;
    if (i < MPAD)      ws[WS_ENERGY + i] = 0.0f;
    if (i < 12 * 256)  ws[WS_FACTS + MCL_DIM * E_DIM + i] = 0.0f;   // pad rows 500..511
    if (i < F_DIM)     ws[WS_X + E_DIM + i] = 0.0f;                 // decoder positional zeros
}

// ---------------- encoder: facts[c, e] = sum_f word_emb[context[c,f], e] * l(f,e) ---------
__global__ void k_encode(const int* __restrict__ context, const float* __restrict__ word_emb,
                         float* __restrict__ ws) {
    __shared__ int toks[F_DIM];
    int c = blockIdx.x;            // 0..498
    int e = threadIdx.x;           // 0..255
    if (e < F_DIM) toks[e] = context[c * F_DIM + e];
    __syncthreads();
    float en  = (float)e / 255.0f;
    float acc = 0.0f;
    for (int f = 0; f < F_DIM; ++f) {
        float s = (float)f / 99.0f;
        float l = 1.0f - s - en * (1.0f - 2.0f * s);
        acc += word_emb[(long long)toks[f] * E_DIM + e] * l;
    }
    ws[WS_FACTS + c * E_DIM + e] = acc;
}

// ---------------- mean fact -> row 499 ----------------
__global__ void k_meanfact(float* __restrict__ ws, const int* __restrict__ nf) {
    int e = threadIdx.x;           // 0..255
    int n = nf[0];
    float acc = 0.0f;
    for (int c = 0; c < n; ++c) acc += ws[WS_FACTS + c * E_DIM + e];
    ws[WS_FACTS + (MCL_DIM - 1) * E_DIM + e] = acc / (float)n;
}

// ---------------- hterm[j] = attn1_b[j] + h0 . attn1_W[j, 256:768] ----------------
__global__ void k_hterm(const float* __restrict__ attn1_W, const float* __restrict__ attn1_b,
                        const float* __restrict__ h0, float* __restrict__ ws) {
    int j = blockIdx.x;            // 0..511
    int t = threadIdx.x;           // 0..255
    float acc = 0.0f;
    if (j < MCL_DIM)
        for (int k = t; k < H_DIM; k += 256)
            acc += h0[k] * attn1_W[j * (E_DIM + H_DIM) + E_DIM + k];
    __shared__ float red[256];
    red[t] = acc; __syncthreads();
    for (int s = 128; s > 0; s >>= 1) { if (t < s) red[t] += red[t + s]; __syncthreads(); }
    if (t == 0) ws[WS_HTERM + j] = (j < MCL_DIM) ? red[0] + attn1_b[j] : 0.0f;
}

// ---------------- WMMA energy: energy[i] = sum_j tanh(facts_i . W1_j + hterm[j]) * w2[j] --
// Tiles of 16x16 via V_WMMA_F32_16X16X4_F32, K = 256, one wave per tile, 32x32 tiles.
__global__ void k_energy(float* __restrict__ ws) {
    const float* factsP = ws + WS_FACTS;
    const float* w1P    = ws + WS_W1P;
    const float* hterm  = ws + WS_HTERM;
    const float* w2     = ws + WS_W2P;
    float*       energy = ws + WS_ENERGY;

    int wid  = (blockIdx.x * blockDim.x + threadIdx.x) >> 5;  // 0..1023
    int lane = threadIdx.x & 31;
    int tm = wid >> 5, tn = wid & 31;                         // tile coords, 32x32
    int half = lane >> 4;                                     // 0: lanes 0-15, 1: lanes 16-31
    int l16  = lane & 15;

    const float* arow = factsP + (tm * 16 + l16) * E_DIM;     // A: M = tm*16 + l16
    const float* brow = w1P    + (tn * 16 + l16) * E_DIM;     // B: N = tn*16 + l16 (K along row)

    v8f c = {};
    for (int k = 0; k < E_DIM; k += 4) {
        v2f a, b;
        // A 16x4 fp32 layout: VGPR0/1 = K={0,1} (lanes 0-15), K={2,3} (lanes 16-31)
        a.x = arow[k + 2 * half];
        a.y = arow[k + 2 * half + 1];
        // B 4x16 fp32 layout mirrors: rows K striped across lane halves
        b.x = brow[k + 2 * half];
        b.y = brow[k + 2 * half + 1];
        c = __builtin_amdgcn_wmma_f32_16x16x4_f32(false, a, false, b, (short)0, c,
                                                  false, false);
    }

    // C layout: VGPR v -> M = v + 8*half, N = lane&15. Fuse tanh, attn2 weight, row-reduce.
    #pragma unroll
    for (int v = 0; v < 8; ++v) {
        int M = v + 8 * half;
        int j = tn * 16 + l16;                                // global column
        float e = tanhf(c[v] + hterm[j]) * w2[j];
        #pragma unroll
        for (int s = 1; s < 16; s <<= 1) e += __shfl_xor(e, s, 32);
        if (l16 == 0) atomicAdd(&energy[tm * 16 + M], e);
    }
}

// ---------------- softmax + mask + renorm + log_aw + argmax + attn_weighted + build x ------
__global__ void k_attention(float* __restrict__ ws, const float* __restrict__ word_emb,
                            const float* __restrict__ attn2_b, const int* __restrict__ nf,
                            float* __restrict__ d_out) {
    __shared__ float sv[512];
    __shared__ int   si[512];
    __shared__ float s_aw[512];
    int t = threadIdx.x;           // 0..511
    int n = nf[0];

    float ev = (t < MCL_DIM) ? ws[WS_ENERGY + t] + attn2_b[0] : -3.4e38f;
    sv[t] = ev; __syncthreads();
    for (int s = 256; s > 0; s >>= 1) { if (t < s) sv[t] = fmaxf(sv[t], sv[t + s]); __syncthreads(); }
    float m = sv[0]; __syncthreads();

    float ex = (t < MCL_DIM) ? expf(ev - m) : 0.0f;
    sv[t] = ex; __syncthreads();
    for (int s = 256; s > 0; s >>= 1) { if (t < s) sv[t] += sv[t + s]; __syncthreads(); }
    float S = sv[0]; __syncthreads();

    float aw   = ex / S;
    float mask = ((t < n) || (t == MCL_DIM - 1)) ? 1.0f : 0.0f;
    float awm  = (t < MCL_DIM) ? aw * mask + EPS_F : 0.0f;
    sv[t] = awm; __syncthreads();
    for (int s = 256; s > 0; s >>= 1) { if (t < s) sv[t] += sv[t + s]; __syncthreads(); }
    float S2 = sv[0]; __syncthreads();

    awm = awm / S2;
    s_aw[t] = (t < MCL_DIM) ? awm : 0.0f;
    if (t < MCL_DIM) d_out[V_DIM + t] = logf(awm);            // log_aw output

    // argmax (first occurrence of max)
    sv[t] = (t < MCL_DIM) ? awm : -1.0f; si[t] = t; __syncthreads();
    for (int s = 256; s > 0; s >>= 1) {
        if (t < s) {
            if (sv[t + s] > sv[t] || (sv[t + s] == sv[t] && si[t + s] < si[t])) {
                sv[t] = sv[t + s]; si[t] = si[t + s];
            }
        }
        __syncthreads();
    }
    int fmax = si[0];
    __syncthreads();

    if (t < E_DIM) {
        float acc = 0.0f;
        for (int j = 0; j < MCL_DIM; ++j) acc += s_aw[j] * ws[WS_FACTS + j * E_DIM + t];
        ws[WS_ATTNW + t] = acc;                               // attn_weighted_facts
        ws[WS_X + t] = word_emb[E_DIM + t];                   // word_emb[1]
        ws[WS_X + E_DIM + F_DIM + t] = ws[WS_FACTS + fmax * E_DIM + t];  // emb_f_max
    }
}

// ---------------- GRU gate GEMVs: gi = W_ih x + b_ih ; gh = W_hh h0 + b_hh (wave/row) -----
__global__ void k_gates(const float* __restrict__ W_ih, const float* __restrict__ b_ih,
                        const float* __restrict__ W_hh, const float* __restrict__ b_hh,
                        const float* __restrict__ h0, float* __restrict__ ws) {
    int wid  = (blockIdx.x * blockDim.x + threadIdx.x) >> 5;  // 0..3071
    int lane = threadIdx.x & 31;
    const int XK = 2 * E_DIM + F_DIM;                         // 612
    float acc = 0.0f;
    if (wid < 3 * H_DIM) {
        for (int k = lane; k < XK; k += 32) acc += ws[WS_X + k] * W_ih[wid * XK + k];
        #pragma unroll
        for (int s = 1; s < 32; s <<= 1) acc += __shfl_xor(acc, s, 32);
        if (lane == 0) ws[WS_GI + wid] = acc + b_ih[wid];
    } else {
        int r = wid - 3 * H_DIM;
        for (int k = lane; k < H_DIM; k += 32) acc += h0[k] * W_hh[r * H_DIM + k];
        #pragma unroll
        for (int s = 1; s < 32; s <<= 1) acc += __shfl_xor(acc, s, 32);
        if (lane == 0) ws[WS_GH + r] = acc + b_hh[r];
    }
}

// ---------------- GRU cell combine + outvec = [h_new | attn_weighted] ----------------
__global__ void k_gru(const float* __restrict__ h0, float* __restrict__ ws,
                      float* __restrict__ d_out) {
    int t = threadIdx.x;           // 0..511
    float r = 1.0f / (1.0f + expf(-(ws[WS_GI + t]             + ws[WS_GH + t])));
    float z = 1.0f / (1.0f + expf(-(ws[WS_GI + H_DIM + t]     + ws[WS_GH + H_DIM + t])));
    float nn = tanhf(ws[WS_GI + 2 * H_DIM + t] + r * ws[WS_GH + 2 * H_DIM + t]);
    float hn = (1.0f - z) * nn + z * h0[t];
    ws[WS_OUTV + t] = hn;
    d_out[V_DIM + MCL_DIM + t] = hn;                          // decoder_hidden output
    if (t < E_DIM) ws[WS_OUTV + H_DIM + t] = ws[WS_ATTNW + t];
}

// ---------------- out1 = relu(outvec @ out1_W.T + b) (wave per row, K=768) ----------------
__global__ void k_out1(const float* __restrict__ out1_W, const float* __restrict__ out1_b,
                       float* __restrict__ ws) {
    int wid  = (blockIdx.x * blockDim.x + threadIdx.x) >> 5;  // 0..511
    int lane = threadIdx.x & 31;
    const int K = H_DIM + E_DIM;                              // 768
    float acc = 0.0f;
    for (int k = lane; k < K; k += 32) acc += ws[WS_OUTV + k] * out1_W[wid * K + k];
    #pragma unroll
    for (int s = 1; s < 32; s <<= 1) acc += __shfl_xor(acc, s, 32);
    if (lane == 0) ws[WS_OUT1 + wid] = fmaxf(acc + out1_b[wid], 0.0f);
}

// ---------------- vocab head GEMV: 128000 rows x K=512, wave per row, b128 loads ----------
__global__ void k_logits(const float* __restrict__ out2_W, const float* __restrict__ out2_b,
                         const float* __restrict__ ws, float* __restrict__ d_out) {
    int wid  = (blockIdx.x * blockDim.x + threadIdx.x) >> 5;  // vocab row
    int lane = threadIdx.x & 31;
    const float4* w = (const float4*)(out2_W + (size_t)wid * H_DIM);
    const float4* u = (const float4*)(ws + WS_OUT1);
    float acc = 0.0f;
    #pragma unroll
    for (int i = lane; i < H_DIM / 4; i += 32) {
        float4 a = w[i], b = u[i];
        acc += a.x * b.x + a.y * b.y + a.z * b.z + a.w * b.w;
    }
    #pragma unroll
    for (int s = 1; s < 32; s <<= 1) acc += __shfl_xor(acc, s, 32);
    if (lane == 0) d_out[wid] = acc + out2_b[wid];
}

// ---------------- log_softmax over V in place ----------------
__global__ void k_logsoftmax(float* __restrict__ d_out) {
    __shared__ float red[1024];
    int t = threadIdx.x;
    float m = -3.4e38f;
    for (int i = t; i < V_DIM; i += 1024) m = fmaxf(m, d_out[i]);
    red[t] = m; __syncthreads();
    for (int s = 512; s > 0; s >>= 1) { if (t < s) red[t] = fmaxf(red[t], red[t + s]); __syncthreads(); }
    m = red[0]; __syncthreads();
    float sum = 0.0f;
    for (int i = t; i < V_DIM; i += 1024) sum += expf(d_out[i] - m);
    red[t] = sum; __syncthreads();
    for (int s = 512; s > 0; s >>= 1) { if (t < s) red[t] += red[t + s]; __syncthreads(); }
    float lse = m + logf(red[0]);
    __syncthreads();
    for (int i = t; i < V_DIM; i += 1024) d_out[i] -= lse;
}

// ---------------- launch ----------------
extern "C" void kernel_launch(void* const* d_in, const int* in_sizes, int n_in,
                              void* d_out_v, int out_size, void* d_ws, size_t ws_size,
                              hipStream_t stream) {
    const int*   context = (const int*)  d_in[0];
    const float* word_emb= (const float*)d_in[1];
    const float* attn1_W = (const float*)d_in[2];
    const float* attn1_b = (const float*)d_in[3];
    const float* attn2_W = (const float*)d_in[4];
    const float* attn2_b = (const float*)d_in[5];
    const float* W_ih    = (const float*)d_in[6];
    const float* W_hh    = (const float*)d_in[7];
    const float* b_ih    = (const float*)d_in[8];
    const float* b_hh    = (const float*)d_in[9];
    const float* out1_W  = (const float*)d_in[10];
    const float* out1_b  = (const float*)d_in[11];
    const float* out2_W  = (const float*)d_in[12];
    const float* out2_b  = (const float*)d_in[13];
    const float* h0      = (const float*)d_in[14];
    const int*   nf      = (const int*)  d_in[15];
    float* d_out = (float*)d_out_v;
    float* ws    = (float*)d_ws;

    k_prep      <<<MPAD, 256, 0, stream>>>(attn1_W, attn2_W, ws);         // 512*256 threads
    k_encode    <<<CTX_DIM, 256, 0, stream>>>(context, word_emb, ws);
    k_meanfact  <<<1, 256, 0, stream>>>(ws, nf);
    k_hterm     <<<MPAD, 256, 0, stream>>>(attn1_W, attn1_b, h0, ws);
    k_energy    <<<128, 256, 0, stream>>>(ws);                            // 1024 waves = 32x32 tiles
    k_attention <<<1, 512, 0, stream>>>(ws, word_emb, attn2_b, nf, d_out);
    k_gates     <<<384, 256, 0, stream>>>(W_ih, b_ih, W_hh, b_hh, h0, ws);
    k_gru       <<<1, 512, 0, stream>>>(h0, ws, d_out);
    k_out1      <<<64, 256, 0, stream>>>(out1_W, out1_b, ws);
    k_logits    <<<V_DIM / 8, 256, 0, stream>>>(out2_W, out2_b, ws, d_out);
    k_logsoftmax<<<1, 1024, 0, stream>>>(d_out);
}